// GMP_11673721111143
// MI455X (gfx1250) — compile-verified
//
#include <hip/hip_runtime.h>

typedef __attribute__((ext_vector_type(2))) float v2f;
typedef __attribute__((ext_vector_type(8))) float v8f;

#define TS 128        // output samples per block
#define HALO 16
#define R   (TS + 2*HALO)   // 160 LDS rows: [blockStart-16, blockStart+TS+16)
#define NCOL 144      // columns of W/G: a_re(8) a_im(8) b_re(32) b_im(32) c_re(32) c_im(32)
#define GSTR 145      // padded G row stride in floats (odd -> conflict-free gathers)
#define PSTR 9        // padded P row stride
#define NTHREADS 128
#define NWAVES (NTHREADS/32)
#define ROWTILES (R/16)     // 10
#define COLTILES (NCOL/16)  // 9

__global__ __launch_bounds__(NTHREADS)
void gmp_wmma_kernel(const float* __restrict__ x,
                     const float* __restrict__ a_re, const float* __restrict__ a_im,
                     const float* __restrict__ b_re, const float* __restrict__ b_im,
                     const float* __restrict__ c_re, const float* __restrict__ c_im,
                     float* __restrict__ out, int N)
{
    __shared__ float W[8][NCOL];        // 4.5 KB  coefficient matrix (8 x 144)
    __shared__ float P[R][PSTR];        // 5.6 KB  power vectors (halo'd tile)
    __shared__ float XS[R][2];          // 1.3 KB  complex samples (halo'd tile)
    __shared__ float G[R * GSTR];       // 90.6 KB G = P @ W

    const int tid = threadIdx.x;
    const int blockStart = blockIdx.x * TS;

    // ---- Phase 0: build coefficient matrix W[k][col] ----
    for (int idx = tid; idx < 8 * NCOL; idx += NTHREADS) {
        const int k = idx / NCOL, col = idx % NCOL;
        float v;
        if (col < 8)        v = a_re[k * 8  + col];          // a[k][l], l=col
        else if (col < 16)  v = a_im[k * 8  + (col - 8)];
        else if (col < 48)  v = b_re[k * 32 + (col - 16)];   // b[k][l][m], l*4+m = col-16
        else if (col < 80)  v = b_im[k * 32 + (col - 48)];
        else if (col < 112) v = c_re[k * 32 + (col - 80)];
        else                v = c_im[k * 32 + (col - 112)];
        W[k][col] = v;
    }

    // ---- Phase 1: magnitudes -> power vectors P[j][k] = |x_j|^k, plus XS ----
    for (int j = tid; j < R; j += NTHREADS) {
        int g = blockStart + j - HALO;
        g = min(max(g, 0), N - 1);                 // matches reference jnp.clip
        const float xr = x[2 * g], xi = x[2 * g + 1];
        XS[j][0] = xr; XS[j][1] = xi;
        const float r = sqrtf(xr * xr + xi * xi);
        float p = 1.0f;
        #pragma unroll
        for (int k = 0; k < 8; ++k) { P[j][k] = p; p *= r; }
    }
    __syncthreads();

    // ---- Phase 2: G(160x144) = P(160x8) @ W(8x144) via v_wmma_f32_16x16x4_f32 ----
    // ISA f32 layouts: A: lane&15 = M row, VGPR g holds K=g (lanes 0-15) / K=g+2 (16-31)
    //                  B: VGPR g holds K row g / g+2, lane&15 = N col
    //                  D: VGPR i holds M = i + 8*(lane>>4), N = lane&15
    {
        const int wave  = tid >> 5;
        const int lane  = tid & 31;
        const int mrow  = lane & 15;
        const int khalf = (lane >> 4) * 2;         // 0 or 2
        for (int job = wave; job < ROWTILES * COLTILES; job += NWAVES) {
            const int rt = (job / COLTILES) * 16;
            const int ct = (job % COLTILES) * 16;
            v8f acc = {};
            #pragma unroll
            for (int ks = 0; ks < 8; ks += 4) {    // K=8 as two chained K=4 steps
                v2f Af, Bf;
                Af.x = P[rt + mrow][ks + khalf + 0];
                Af.y = P[rt + mrow][ks + khalf + 1];
                Bf.x = W[ks + khalf + 0][ct + mrow];
                Bf.y = W[ks + khalf + 1][ct + mrow];
                acc = __builtin_amdgcn_wmma_f32_16x16x4_f32(
                    false, Af, false, Bf, (short)0, acc, false, false);
            }
            const int rowBase = rt + (lane >> 4) * 8;
            #pragma unroll
            for (int i = 0; i < 8; ++i)
                G[(rowBase + i) * GSTR + ct + mrow] = acc[i];
        }
    }
    __syncthreads();

    // ---- Phase 3: combine  y[n] = sum_l x[n-l] * (A + sum_m B + sum_m C) ----
    {
        const int t = tid;                          // one output per thread
        float accRe = 0.f, accIm = 0.f;
        #pragma unroll
        for (int l = 0; l < 8; ++l) {
            const int r0 = t + HALO - l;            // LDS row of sample n-l
            const float* gr = &G[r0 * GSTR];
            float sRe = gr[l];                      // aligned term a
            float sIm = gr[8 + l];
            #pragma unroll
            for (int m = 0; m < 4; ++m) {
                const float* gb = &G[(r0 - m) * GSTR];   // envelope lag  (n-l-m)
                sRe += gb[16 + l * 4 + m];
                sIm += gb[48 + l * 4 + m];
                const float* gc = &G[(r0 + m) * GSTR];   // envelope lead (n-l+m)
                sRe += gc[80  + l * 4 + m];
                sIm += gc[112 + l * 4 + m];
            }
            const float xr = XS[r0][0], xi = XS[r0][1];
            accRe += xr * sRe - xi * sIm;           // complex x[n-l] * gain_l
            accIm += xr * sIm + xi * sRe;
        }
        const int n = blockStart + t;
        if (n < N) { out[2 * n] = accRe; out[2 * n + 1] = accIm; }
    }
}

extern "C" void kernel_launch(void* const* d_in, const int* in_sizes, int n_in,
                              void* d_out, int out_size, void* d_ws, size_t ws_size,
                              hipStream_t stream) {
    (void)n_in; (void)out_size; (void)d_ws; (void)ws_size;
    const float* x    = (const float*)d_in[0];
    const float* a_re = (const float*)d_in[1];
    const float* a_im = (const float*)d_in[2];
    const float* b_re = (const float*)d_in[3];
    const float* b_im = (const float*)d_in[4];
    const float* c_re = (const float*)d_in[5];
    const float* c_im = (const float*)d_in[6];
    float* out = (float*)d_out;
    const int N = in_sizes[0] / 2;                  // x is (N,2)
    const int grid = (N + TS - 1) / TS;
    gmp_wmma_kernel<<<grid, NTHREADS, 0, stream>>>(x, a_re, a_im, b_re, b_im,
                                                   c_re, c_im, out, N);
}